// OHEMBCEWithLogitsLoss_37847251812781
// MI455X (gfx1250) — compile-verified
//
#include <hip/hip_runtime.h>
#include <hip/hip_bf16.h>
#include <math.h>

// OHEM BCE-with-logits via exact radix-select, TWO streaming passes.
// Key insight: for negatives (t=0) loss = softplus(x) is strictly monotonic
// in x, so top-k by loss == top-k by order-preserving float key. After a
// 13-bit MSB histogram selects the threshold bin, the remaining 19 key bits
// are histogrammed exactly (2^19 bins, global), so every residual bin is an
// EXACT float value and the tail sum is computed from counts alone — no
// third pass over the 226MB inputs. Roofline: 2 x 226MB / 23.3TB/s ~ 19.5us.

#define RADIX_BITS 13
#define NBINS (1 << RADIX_BITS)          // 8192 bins -> 32KB LDS histogram
#define LOW_BITS (32 - RADIX_BITS)       // 19 residual bits
#define NLOW (1 << LOW_BITS)             // 524288 bins -> 2MB global histogram
#define BLK 256                          // 8 wave32 waves
#define NPASS_BLOCKS 512
#define NCHUNK 256                       // must equal BLK
#define CHUNK (NLOW / NCHUNK)            // 2048 bins per chunk

typedef __attribute__((ext_vector_type(2))) float v2f;
typedef __attribute__((ext_vector_type(8))) float v8f;

struct Ws {
  unsigned long long pos_sum_fx;   // fixed point 32.32 (deterministic)
  unsigned long long neg_sum_fx;   // fixed point 32.32
  unsigned int n_pos;
  unsigned int k;
  unsigned int bin1;               // selected top-13-bit bin
  unsigned int cnt_above1;         // # neg keys with top13 > bin1 (strict, < k)
  unsigned int tbin;               // selected low-19-bit bin (exact key)
  unsigned int ties;               // # elements taken at the threshold key
  unsigned int chunksum[NCHUNK];
  unsigned int hist1[NBINS];
  unsigned int hist2[NLOW];
};  // ~2.13 MB of scratch required

__device__ __forceinline__ float bce_loss(float x, float t) {
  return fmaxf(x, 0.0f) - x * t + log1pf(expf(-fabsf(x)));
}

// order-preserving key: larger float -> larger unsigned key
__device__ __forceinline__ unsigned int sort_key(float x) {
  unsigned int u = __float_as_uint(x);
  return u ^ ((u & 0x80000000u) ? 0xFFFFFFFFu : 0x80000000u);
}

__device__ __forceinline__ float key_to_float(unsigned int key) {
  unsigned int u = key ^ ((key & 0x80000000u) ? 0x80000000u : 0xFFFFFFFFu);
  return __uint_as_float(u);
}

// Block reduction of two f32 values using WMMA in a layout-agnostic form:
// A = all-ones 16x4 (layout irrelevant: every element is 1.0), B holds one
// partial per lane (other slot 0), so D[0][n] are column sums and
// sum_n D[0][n] == sum of all lane partials. D row0 layout is documented:
// result VGPR0, lanes 0..15. Results valid on threadIdx.x==0.
__device__ __forceinline__ void block_reduce2(float a, float b, float* oa, float* ob) {
  __shared__ float red[2][BLK / 32][16];
  v2f A; A.x = 1.0f; A.y = 1.0f;
  v2f Ba; Ba.x = a; Ba.y = 0.0f;
  v2f Bb; Bb.x = b; Bb.y = 0.0f;
  v8f Ca = {};
  v8f Cb = {};
  Ca = __builtin_amdgcn_wmma_f32_16x16x4_f32(false, A, false, Ba, (short)0, Ca, false, false);
  Cb = __builtin_amdgcn_wmma_f32_16x16x4_f32(false, A, false, Bb, (short)0, Cb, false, false);
  int lane = threadIdx.x & 31;
  int wid  = threadIdx.x >> 5;
  if (lane < 16) { red[0][wid][lane] = Ca[0]; red[1][wid][lane] = Cb[0]; }
  __syncthreads();
  if (threadIdx.x == 0) {
    float sa = 0.0f, sb = 0.0f;
    for (int w = 0; w < BLK / 32; ++w)
      for (int i = 0; i < 16; ++i) { sa += red[0][w][i]; sb += red[1][w][i]; }
    *oa = sa; *ob = sb;
  }
}

__global__ void k_init(unsigned int* w, int nwords) {
  int i = blockIdx.x * blockDim.x + threadIdx.x;
  if (i < nwords) w[i] = 0u;
}

// Pass 1: positive loss sum + count; 13-bit MSB histogram of negative keys.
__global__ __launch_bounds__(BLK)
void k_pass1(const float* __restrict__ lg, const float* __restrict__ tg,
             unsigned long long n, Ws* __restrict__ ws) {
  __shared__ unsigned int lh[NBINS];
  for (int i = threadIdx.x; i < NBINS; i += BLK) lh[i] = 0u;
  __syncthreads();
  unsigned long long n4 = n >> 2;
  const float4* l4 = (const float4*)lg;
  const float4* t4 = (const float4*)tg;
  float pos = 0.0f, np = 0.0f;
  unsigned long long stride = (unsigned long long)gridDim.x * BLK;
  for (unsigned long long i = (unsigned long long)blockIdx.x * BLK + threadIdx.x;
       i < n4; i += stride) {
    __builtin_prefetch(&l4[i + 16ull * stride], 0, 1);
    __builtin_prefetch(&t4[i + 16ull * stride], 0, 1);
    float4 xv = l4[i];
    float4 tv = t4[i];
#define P1(X, T) do { float x_ = (X), t_ = (T);                                  \
      if (t_ > 0.5f) { pos += bce_loss(x_, t_); np += 1.0f; }                    \
      else atomicAdd(&lh[sort_key(x_) >> LOW_BITS], 1u); } while (0)
    P1(xv.x, tv.x); P1(xv.y, tv.y); P1(xv.z, tv.z); P1(xv.w, tv.w);
#undef P1
  }
  unsigned int rem = (unsigned int)(n & 3ull);
  if (blockIdx.x == 0 && threadIdx.x < rem) {
    float x = lg[(n4 << 2) + threadIdx.x];
    float t = tg[(n4 << 2) + threadIdx.x];
    if (t > 0.5f) { pos += bce_loss(x, t); np += 1.0f; }
    else atomicAdd(&lh[sort_key(x) >> LOW_BITS], 1u);
  }
  __syncthreads();
  for (int i = threadIdx.x; i < NBINS; i += BLK) {
    unsigned int c = lh[i];
    if (c) atomicAdd(&ws->hist1[i], c);
  }
  float bp, bn;
  block_reduce2(pos, np, &bp, &bn);
  if (threadIdx.x == 0) {
    atomicAdd(&ws->pos_sum_fx, (unsigned long long)((double)bp * 4294967296.0 + 0.5));
    atomicAdd(&ws->n_pos, (unsigned int)(bn + 0.5f));
  }
}

// Select the top-13-bit bin containing the k-th largest negative key.
__global__ __launch_bounds__(BLK)
void k_select1(Ws* __restrict__ ws, unsigned long long n_total) {
  __shared__ unsigned int csum[BLK];
  const int per = NBINS / BLK;  // 32 bins per thread
  int t = threadIdx.x;
  unsigned int s = 0;
  for (int j = 0; j < per; ++j) s += ws->hist1[t * per + j];
  csum[t] = s;
  __syncthreads();
  if (t == 0) {
    unsigned int n_pos = ws->n_pos;
    long long n_neg = (long long)n_total - (long long)n_pos;
    long long k = 0;
    if (n_neg > 0) {
      float kf = rintf(0.1f * (float)n_neg);  // f32 mul + RNE, matches jnp.round
      k = (long long)kf;
      if (k < 1024) k = 1024;
      if (k > n_neg) k = n_neg;
    }
    ws->k = (unsigned int)k;
    unsigned int cum = 0;
    int chunk = -1;
    if (k > 0) {
      for (int c = BLK - 1; c >= 0; --c) {
        if ((unsigned long long)cum + csum[c] >= (unsigned long long)k) { chunk = c; break; }
        cum += csum[c];
      }
    }
    unsigned int bin1 = 0;
    unsigned int above = cum;
    if (chunk >= 0) {
      for (int b = chunk * per + per - 1; b >= chunk * per; --b) {
        unsigned int h = ws->hist1[b];
        if ((unsigned long long)above + h >= (unsigned long long)k) { bin1 = (unsigned int)b; break; }
        above += h;
      }
    }
    ws->bin1 = bin1;
    ws->cnt_above1 = above;   // strictly above bin1, guaranteed < k
  }
}

// Pass 2 (fused): negatives with top13 > bin1 are guaranteed top-k members ->
// sum their losses now. Negatives with top13 == bin1 go into the exact
// 19-bit residual histogram (global, sparse traffic).
__global__ __launch_bounds__(BLK)
void k_pass2(const float* __restrict__ lg, const float* __restrict__ tg,
             unsigned long long n, Ws* __restrict__ ws) {
  unsigned int bin1 = ws->bin1;
  unsigned int kk = ws->k;
  float neg = 0.0f;
  if (kk != 0) {
    unsigned long long n4 = n >> 2;
    const float4* l4 = (const float4*)lg;
    const float4* t4 = (const float4*)tg;
    unsigned long long stride = (unsigned long long)gridDim.x * BLK;
    for (unsigned long long i = (unsigned long long)blockIdx.x * BLK + threadIdx.x;
         i < n4; i += stride) {
      __builtin_prefetch(&l4[i + 16ull * stride], 0, 1);
      __builtin_prefetch(&t4[i + 16ull * stride], 0, 1);
      float4 xv = l4[i];
      float4 tv = t4[i];
#define P2(X, T) do { float x_ = (X), t_ = (T);                                  \
        if (!(t_ > 0.5f)) {                                                      \
          unsigned int key = sort_key(x_);                                       \
          unsigned int top = key >> LOW_BITS;                                    \
          if (top > bin1) neg += bce_loss(x_, t_);                               \
          else if (top == bin1) atomicAdd(&ws->hist2[key & (NLOW - 1)], 1u);     \
        } } while (0)
      P2(xv.x, tv.x); P2(xv.y, tv.y); P2(xv.z, tv.z); P2(xv.w, tv.w);
#undef P2
    }
    unsigned int rem = (unsigned int)(n & 3ull);
    if (blockIdx.x == 0 && threadIdx.x < rem) {
      float x = lg[(n4 << 2) + threadIdx.x];
      float t = tg[(n4 << 2) + threadIdx.x];
      if (!(t > 0.5f)) {
        unsigned int key = sort_key(x);
        unsigned int top = key >> LOW_BITS;
        if (top > bin1) neg += bce_loss(x, t);
        else if (top == bin1) atomicAdd(&ws->hist2[key & (NLOW - 1)], 1u);
      }
    }
  }
  float bn, dummy;
  block_reduce2(neg, 0.0f, &bn, &dummy);
  if (threadIdx.x == 0) {
    atomicAdd(&ws->neg_sum_fx, (unsigned long long)((double)bn * 4294967296.0 + 0.5));
  }
}

// Per-chunk sums of the 19-bit residual histogram (256 blocks x 2048 bins).
__global__ __launch_bounds__(BLK)
void k_chunks(Ws* __restrict__ ws) {
  __shared__ unsigned int sh[BLK];
  unsigned int base = blockIdx.x * CHUNK;
  unsigned int s = 0;
  for (int j = threadIdx.x; j < CHUNK; j += BLK) s += ws->hist2[base + j];
  sh[threadIdx.x] = s;
  __syncthreads();
  for (int st = BLK / 2; st > 0; st >>= 1) {
    if ((int)threadIdx.x < st) sh[threadIdx.x] += sh[threadIdx.x + st];
    __syncthreads();
  }
  if (threadIdx.x == 0) ws->chunksum[blockIdx.x] = sh[0];
}

// Find the exact threshold key within bin1 and the tie count.
__global__ __launch_bounds__(BLK)
void k_findbin(Ws* __restrict__ ws) {
  __shared__ unsigned int cs[NCHUNK];
  __shared__ unsigned int binbuf[CHUNK];
  __shared__ unsigned int sel[2];  // selected chunk, cum strictly above it
  int t = threadIdx.x;
  cs[t] = ws->chunksum[t];
  __syncthreads();
  if (t == 0) {
    unsigned int k = ws->k;
    sel[0] = 0xFFFFFFFFu;
    sel[1] = 0u;
    if (k) {
      unsigned int cum = ws->cnt_above1;
      for (int c = NCHUNK - 1; c >= 0; --c) {
        if ((unsigned long long)cum + cs[c] >= (unsigned long long)k) {
          sel[0] = (unsigned int)c; sel[1] = cum; break;
        }
        cum += cs[c];
      }
    }
  }
  __syncthreads();
  unsigned int chunk = sel[0];
  if (chunk == 0xFFFFFFFFu) {           // k == 0 (no negatives)
    if (t == 0) { ws->tbin = 0u; ws->ties = 0u; }
    return;
  }
  for (int j = t; j < CHUNK; j += BLK) binbuf[j] = ws->hist2[chunk * CHUNK + j];
  __syncthreads();
  if (t == 0) {
    unsigned int k = ws->k;
    unsigned int cum = sel[1];
    unsigned int tbin = chunk * CHUNK;
    unsigned int ties = 0;
    for (int j = CHUNK - 1; j >= 0; --j) {
      unsigned int h = binbuf[j];
      if ((unsigned long long)cum + h >= (unsigned long long)k) {
        tbin = chunk * CHUNK + (unsigned int)j;
        ties = k - cum;                  // 1..h, exact
        break;
      }
      cum += h;
    }
    ws->tbin = tbin;
    ws->ties = ties;
  }
}

// Sum count*softplus(value) over residual bins strictly above the threshold.
// Every bin is an exact 32-bit key -> exact float -> exact contribution.
// Fixed-point per-bin conversion + integer adds: bitwise deterministic.
__global__ __launch_bounds__(BLK)
void k_sumabove(Ws* __restrict__ ws) {
  __shared__ unsigned long long sh[BLK];
  unsigned int kk = ws->k;
  unsigned int tb = ws->tbin;
  unsigned int b1 = ws->bin1;
  unsigned int base = blockIdx.x * CHUNK;
  unsigned long long part = 0ull;
  if (kk) {
    for (int j = threadIdx.x; j < CHUNK; j += BLK) {
      unsigned int idx = base + (unsigned int)j;
      if (idx > tb) {
        unsigned int c = ws->hist2[idx];
        if (c) {
          float x = key_to_float((b1 << LOW_BITS) | idx);
          float l = fmaxf(x, 0.0f) + log1pf(expf(-fabsf(x)));  // t == 0
          part += (unsigned long long)((double)l * 4294967296.0 + 0.5) *
                  (unsigned long long)c;
        }
      }
    }
  }
  sh[threadIdx.x] = part;
  __syncthreads();
  for (int st = BLK / 2; st > 0; st >>= 1) {
    if ((int)threadIdx.x < st) sh[threadIdx.x] += sh[threadIdx.x + st];
    __syncthreads();
  }
  if (threadIdx.x == 0 && sh[0]) atomicAdd(&ws->neg_sum_fx, sh[0]);
}

// Finalize: add exact tie contribution, compute means.
__global__ void k_final(const Ws* __restrict__ ws, unsigned long long n_total,
                        float* __restrict__ out) {
  if (threadIdx.x != 0 || blockIdx.x != 0) return;
  unsigned int n_pos = ws->n_pos;
  unsigned long long n_neg = n_total - (unsigned long long)n_pos;
  unsigned int k = ws->k;
  double pos_sum = (double)ws->pos_sum_fx / 4294967296.0;
  double neg_sum = (double)ws->neg_sum_fx / 4294967296.0;
  if (k > 0 && ws->ties > 0) {
    float x = key_to_float((ws->bin1 << LOW_BITS) | ws->tbin);
    float l = fmaxf(x, 0.0f) + log1pf(expf(-fabsf(x)));  // t == 0
    neg_sum += (double)ws->ties * (double)l;
  }
  float pos_mean = (n_pos > 0u) ? (float)(pos_sum / (double)n_pos) : 0.0f;
  float neg_mean = (n_neg > 0ull) ? (float)(neg_sum / (double)(k ? k : 1u)) : 0.0f;
  out[0] = pos_mean + neg_mean;
}

extern "C" void kernel_launch(void* const* d_in, const int* in_sizes, int n_in,
                              void* d_out, int out_size, void* d_ws, size_t ws_size,
                              hipStream_t stream) {
  const float* lg = (const float*)d_in[0];
  const float* tg = (const float*)d_in[1];
  float* out = (float*)d_out;
  unsigned long long n = (unsigned long long)(unsigned int)in_sizes[0];
  Ws* ws = (Ws*)d_ws;  // requires ~2.13MB of scratch
  int nwords = (int)(sizeof(Ws) / 4);
  k_init<<<(nwords + BLK - 1) / BLK, BLK, 0, stream>>>((unsigned int*)d_ws, nwords);
  k_pass1<<<NPASS_BLOCKS, BLK, 0, stream>>>(lg, tg, n, ws);
  k_select1<<<1, BLK, 0, stream>>>(ws, n);
  k_pass2<<<NPASS_BLOCKS, BLK, 0, stream>>>(lg, tg, n, ws);
  k_chunks<<<NCHUNK, BLK, 0, stream>>>(ws);
  k_findbin<<<1, BLK, 0, stream>>>(ws);
  k_sumabove<<<NCHUNK, BLK, 0, stream>>>(ws);
  k_final<<<1, 32, 0, stream>>>(ws, n, out);
}